// RGCN_54443005444391
// MI455X (gfx1250) — compile-verified
//
#include <hip/hip_runtime.h>
#include <hip/hip_bf16.h>

typedef __attribute__((ext_vector_type(2))) float v2f;
typedef __attribute__((ext_vector_type(8))) float v8f;

namespace {
constexpr int N  = 30000;
constexpr int E  = 480000;
constexpr int D  = 128;
constexpr int Hd = 128;
constexpr int B  = 64;
constexpr int NC = 10;
constexpr int R  = 11;
constexpr int H3 = 384;     // 3*H
constexpr int H12 = 1536;   // 12*H
}

// ---------------- degree kernels ----------------
__global__ void deg_count_kernel(const int* __restrict__ src, const int* __restrict__ dst,
                                 float* __restrict__ dout, float* __restrict__ din) {
  int t = blockIdx.x * blockDim.x + threadIdx.x;
  if (t >= R * E) return;
  int r = t / E;
  int s = src[t];
  int d = dst[t];
  __hip_atomic_fetch_add(&dout[r * N + s], 1.0f, __ATOMIC_RELAXED, __HIP_MEMORY_SCOPE_AGENT);
  __hip_atomic_fetch_add(&din [r * N + d], 1.0f, __ATOMIC_RELAXED, __HIP_MEMORY_SCOPE_AGENT);
}

__global__ void deg_finalize_kernel(float* __restrict__ p, int n) {
  int t = blockIdx.x * blockDim.x + threadIdx.x;
  if (t >= n) return;
  p[t] = rsqrtf(fmaxf(p[t], 1.0f));
}

// ---------------- WMMA GEMM: Hout[16-strip] = (X * rs_row) @ W ----------------
// X: [N, K] row-major, W: [K, 128] row-major, Hout: [N, 128]
// grid.x = N/16 blocks of 256 threads (8 waves); wave w owns columns [16w, 16w+16).
__global__ void __launch_bounds__(256)
gemm_rowscale_wmma(const float* __restrict__ X, const float* __restrict__ rs,
                   const float* __restrict__ W, float* __restrict__ Hout, int K) {
  extern __shared__ float sA[];                 // 16 x (K+4)
  const int lstride = K + 4;
  const int mbase = blockIdx.x * 16;
  const int tid = threadIdx.x;

  // stage 16 rows of X, pre-scaled by rsqrt(deg_out)
  for (int r = 0; r < 16; ++r) {
    float s = rs[mbase + r];
    const float* xr = X + (size_t)(mbase + r) * K;
    for (int c = tid; c < K; c += 256)
      sA[r * lstride + c] = xr[c] * s;
  }
  __syncthreads();

  const int lane  = tid & 31;
  const int wave  = tid >> 5;          // 0..7
  const int nbase = wave * 16;         // column tile
  const int m     = lane & 15;         // also the B column offset within tile
  const int khalf = (lane >> 4) * 2;   // 0 or 2

  v8f acc = {};
  for (int kk = 0; kk < K; kk += 4) {
    const int kb = kk + khalf;
    v2f a, b;
    a.x = sA[m * lstride + kb];
    a.y = sA[m * lstride + kb + 1];
    const float* wp = W + (size_t)kb * Hd + nbase + m;
    b.x = wp[0];
    b.y = wp[Hd];
    acc = __builtin_amdgcn_wmma_f32_16x16x4_f32(false, a, false, b,
                                                (short)0, acc, false, false);
  }

  const int col   = nbase + m;
  const int rbase = mbase + (lane >> 4) * 8;
  float* out = Hout + (size_t)rbase * Hd + col;
  #pragma unroll
  for (int j = 0; j < 8; ++j)
    out[(size_t)j * Hd] = acc[j];
}

// ---------------- edge scatter-add: agg[dst] += h[src] ----------------
__global__ void scatter_add_kernel(const float* __restrict__ h, const int* __restrict__ src,
                                   const int* __restrict__ dst, float* __restrict__ agg) {
  int t = blockIdx.x * blockDim.x + threadIdx.x;   // E*128 threads exactly
  int e = t >> 7;
  int c = t & 127;
  float v = h[(size_t)src[e] * Hd + c];
  __hip_atomic_fetch_add(&agg[(size_t)dst[e] * Hd + c], v,
                         __ATOMIC_RELAXED, __HIP_MEMORY_SCOPE_AGENT);
}

// ---------------- combine: out = max(out, relu(agg*rs_in + b)) ----------------
__global__ void combine_kernel(const float* __restrict__ agg, const float* __restrict__ rsin,
                               const float* __restrict__ bias, float* __restrict__ out,
                               int coff) {
  int t = blockIdx.x * blockDim.x + threadIdx.x;   // N*128 threads exactly
  int node = t >> 7;
  int c = t & 127;
  float v = fmaxf(agg[t] * rsin[node] + bias[c], 0.0f);
  float* p = out + (size_t)node * H3 + coff + c;
  *p = fmaxf(*p, v);
}

// ---------------- segment-max pooling over sorted gid ----------------
__global__ void pool_kernel(const float* __restrict__ a0, const float* __restrict__ a1,
                            const float* __restrict__ a2, const float* __restrict__ a3,
                            float* __restrict__ pooled) {
  int t = blockIdx.x * blockDim.x + threadIdx.x;   // B*1536 threads exactly
  int g = t / H12;
  int col = t - g * H12;
  int nt = col / H3;
  int c = col - nt * H3;
  const float* buf = (nt == 0) ? a0 : (nt == 1) ? a1 : (nt == 2) ? a2 : a3;
  // gid[i] = (i*B)//N  =>  graph g covers [ceil(g*N/B), ceil((g+1)*N/B))
  int start = (g * N + B - 1) / B;
  int end   = ((g + 1) * N + B - 1) / B;
  float m = -3.402823e38f;
  for (int i = start; i < end; ++i)
    m = fmaxf(m, buf[(size_t)i * H3 + c]);
  pooled[t] = m;
}

// ---------------- small dense: Y = X @ W + b ----------------
__global__ void linear_kernel(const float* __restrict__ X, const float* __restrict__ W,
                              const float* __restrict__ bias, float* __restrict__ Y,
                              int M, int K, int Nd) {
  int t = blockIdx.x * blockDim.x + threadIdx.x;
  if (t >= M * Nd) return;
  int row = t / Nd;
  int col = t - row * Nd;
  float s = bias[col];
  const float* xr = X + (size_t)row * K;
  for (int k = 0; k < K; ++k)
    s += xr[k] * W[(size_t)k * Nd + col];
  Y[t] = s;
}

// ---------------- training-mode BatchNorm1d + ReLU (in place) ----------------
__global__ void bn_relu_kernel(float* __restrict__ X, const float* __restrict__ g,
                               const float* __restrict__ b, int M, int C) {
  int c = blockIdx.x * blockDim.x + threadIdx.x;
  if (c >= C) return;
  float mu = 0.f;
  for (int i = 0; i < M; ++i) mu += X[(size_t)i * C + c];
  mu /= (float)M;
  float var = 0.f;
  for (int i = 0; i < M; ++i) { float d = X[(size_t)i * C + c] - mu; var += d * d; }
  var /= (float)M;
  float inv = rsqrtf(var + 1e-5f);
  float gc = g[c], bc = b[c];
  for (int i = 0; i < M; ++i) {
    float v = gc * (X[(size_t)i * C + c] - mu) * inv + bc;
    X[(size_t)i * C + c] = fmaxf(v, 0.0f);
  }
}

// ---------------- host orchestration ----------------
extern "C" void kernel_launch(void* const* d_in, const int* in_sizes, int n_in,
                              void* d_out, int out_size, void* d_ws, size_t ws_size,
                              hipStream_t stream) {
  const float* x_attr  = (const float*)d_in[0];
  const float* x_class = (const float*)d_in[1];
  const float* x_type  = (const float*)d_in[2];
  const float* x_val   = (const float*)d_in[3];
  const float* W1      = (const float*)d_in[4];
  const float* b1      = (const float*)d_in[5];
  const float* W2      = (const float*)d_in[6];
  const float* b2      = (const float*)d_in[7];
  const float* fc1_w   = (const float*)d_in[8];
  const float* fc1_b   = (const float*)d_in[9];
  const float* bn1_g   = (const float*)d_in[10];
  const float* bn1_b   = (const float*)d_in[11];
  const float* fc2_w   = (const float*)d_in[12];
  const float* fc2_b   = (const float*)d_in[13];
  const float* bn2_g   = (const float*)d_in[14];
  const float* bn2_b   = (const float*)d_in[15];
  const float* cls_w   = (const float*)d_in[16];
  const float* cls_b   = (const float*)d_in[17];
  const int*   src_idx = (const int*)d_in[18];
  const int*   dst_idx = (const int*)d_in[19];
  float* out = (float*)d_out;

  // workspace carve-up (floats)
  float* ws = (float*)d_ws;
  size_t o = 0;
  float* rs_out = ws + o; o += (size_t)R * N;
  float* rs_in  = ws + o; o += (size_t)R * N;
  float* hbuf   = ws + o; o += (size_t)N * Hd;
  float* agg    = ws + o; o += (size_t)N * Hd;
  const size_t NH3 = (size_t)N * H3;
  float* l1     = ws + o; o += 4 * NH3;
  float* l2     = ws + o; o += 4 * NH3;
  float* pooled = ws + o; o += (size_t)B * H12;
  float* t1     = ws + o; o += (size_t)B * H3;
  float* t2     = ws + o; o += (size_t)B * Hd;

  // degrees (rs_out/rs_in contiguous -> single memset)
  hipMemsetAsync(rs_out, 0, 2 * (size_t)R * N * sizeof(float), stream);
  deg_count_kernel<<<(R * E + 255) / 256, 256, 0, stream>>>(src_idx, dst_idx, rs_out, rs_in);
  deg_finalize_kernel<<<(2 * R * N + 255) / 256, 256, 0, stream>>>(rs_out, 2 * R * N);

  static const int SRC_T[R] = {1, 3, 1, 0, 2, 3, 1, 0, 3, 2, 0};
  static const int DSTB [R] = {1, 1, 0, 0, 0, 0, 2, 2, 2, 2, 3};
  static const int COFF [R] = {0, 256, 128, 128, 128, 256, 128, 128, 128, 256, 256};

  const float* xin0[4] = {x_attr, x_class, x_type, x_val};
  float* l1b[4] = {l1, l1 + NH3, l1 + 2 * NH3, l1 + 3 * NH3};
  float* l2b[4] = {l2, l2 + NH3, l2 + 2 * NH3, l2 + 3 * NH3};

  for (int layer = 0; layer < 2; ++layer) {
    const int K = (layer == 0) ? D : H3;
    const float* Wl = (layer == 0) ? W1 : W2;
    const float* bl = (layer == 0) ? b1 : b2;
    float** outb = (layer == 0) ? l1b : l2b;
    hipMemsetAsync((layer == 0) ? l1 : l2, 0, 4 * NH3 * sizeof(float), stream);

    const size_t smem = 16 * (size_t)(K + 4) * sizeof(float);
    for (int r = 0; r < R; ++r) {
      const float* xr = (layer == 0) ? xin0[SRC_T[r]] : (const float*)l1b[SRC_T[r]];
      gemm_rowscale_wmma<<<N / 16, 256, smem, stream>>>(
          xr, rs_out + (size_t)r * N, Wl + (size_t)r * K * Hd, hbuf, K);
      hipMemsetAsync(agg, 0, (size_t)N * Hd * sizeof(float), stream);
      scatter_add_kernel<<<(size_t)E * Hd / 256, 256, 0, stream>>>(
          hbuf, src_idx + (size_t)r * E, dst_idx + (size_t)r * E, agg);
      combine_kernel<<<(size_t)N * Hd / 256, 256, 0, stream>>>(
          agg, rs_in + (size_t)r * N, bl + (size_t)r * Hd, outb[DSTB[r]], COFF[r]);
    }
  }

  // pooling over graphs (pooled column order: attr | class | type | val)
  pool_kernel<<<(size_t)B * H12 / 256, 256, 0, stream>>>(l2b[0], l2b[1], l2b[2], l2b[3], pooled);

  // MLP head
  linear_kernel<<<(B * H3 + 255) / 256, 256, 0, stream>>>(pooled, fc1_w, fc1_b, t1, B, H12, H3);
  bn_relu_kernel<<<(H3 + 255) / 256, 256, 0, stream>>>(t1, bn1_g, bn1_b, B, H3);
  linear_kernel<<<(B * Hd + 255) / 256, 256, 0, stream>>>(t1, fc2_w, fc2_b, t2, B, H3, Hd);
  bn_relu_kernel<<<(Hd + 255) / 256, 256, 0, stream>>>(t2, bn2_g, bn2_b, B, Hd);
  linear_kernel<<<(B * NC + 255) / 256, 256, 0, stream>>>(t2, cls_w, cls_b, out, B, Hd, NC);
}